// PostProcess_31035433681270
// MI455X (gfx1250) — compile-verified
//
#include <hip/hip_runtime.h>
#include <stdint.h>

// ---------------------------------------------------------------------------
// Pose post-processing for MI455X (gfx1250, wave32).
//   Kernel A: per (b,k) heatmap -> top-30 refined peaks.
//             Heatmap (100KB) staged into LDS via one TDM tensor_load_to_lds.
//   Kernel B: per (b,link) -> 30x30 pair PAF line-integral scores.
//             Both PAF channels (200KB, contiguous) staged to LDS via TDM;
//             per-pair sample reduction via V_WMMA_F32_16X16X4_F32 (full f32).
// ---------------------------------------------------------------------------

#define B_SZ   32
#define K_KP   17
#define HH     160
#define WW     160
#define HW     (HH * WW)        // 25600
#define PP     30
#define SS     10
#define LL     19
#define PEAK_TH 0.1f
#define PAF_TH  0.05f
#define CAND_CAP 8192

typedef __attribute__((ext_vector_type(2)))  float        v2f;
typedef __attribute__((ext_vector_type(8)))  float        v8f;
typedef __attribute__((ext_vector_type(4)))  unsigned int v4u;
typedef __attribute__((ext_vector_type(4)))  int          v4i;
typedef __attribute__((ext_vector_type(8)))  int          v8i;

__constant__ int SK_A[LL] = {15,13,16,14,11, 5, 6, 5, 5, 6, 7, 8, 1, 0, 0, 1, 2, 3, 4};
__constant__ int SK_B[LL] = {13,11,14,12,12,11,12, 6, 7, 8, 9,10, 2, 1, 2, 3, 4, 5, 6};

#if defined(__has_builtin)
#if __has_builtin(__builtin_amdgcn_tensor_load_to_lds) && __has_builtin(__builtin_amdgcn_s_wait_tensorcnt)
#define HAVE_TDM 1
#endif
#endif
#ifndef HAVE_TDM
#define HAVE_TDM 0
#endif

__device__ __forceinline__ unsigned int rfl_u32(unsigned int x) {
  return (unsigned int)__builtin_amdgcn_readfirstlane((int)x);
}

#if HAVE_TDM
// One TDM descriptor: 2-D tensor of 4-byte elements, contiguous rows.
// dims/tiles in elements. Issued by the calling wave; caller must be
// wave-uniform. Waits TENSORcnt==0 before returning.
__device__ __forceinline__ void tdm_load_2d(const void* gptr, unsigned int lds_off,
                                            unsigned int dim0, unsigned int dim1,
                                            unsigned int tile0, unsigned int tile1) {
  unsigned long long gaddr = (unsigned long long)(uintptr_t)gptr;
  // D# group 0: count=1 | lds_addr | global_addr[56:0] | type=2
  v4u g0;
  g0[0] = rfl_u32(1u);
  g0[1] = rfl_u32(lds_off);
  g0[2] = rfl_u32((unsigned int)(gaddr & 0xFFFFFFFFull));
  g0[3] = rfl_u32(((unsigned int)((gaddr >> 32) & 0x1FFFFFFull)) | 0x80000000u);
  // D# group 1: data_size=2 (4B) | tensor_dim0/1 | tile_dim0/1 | dim0 stride
  v8i g1;
  g1[0] = (int)rfl_u32(0x00020000u);
  g1[1] = (int)rfl_u32(dim0 << 16);                                  // dim0 lo16
  g1[2] = (int)rfl_u32(((dim0 >> 16) & 0xFFFFu) | (dim1 << 16));     // dim0 hi | dim1 lo
  g1[3] = (int)rfl_u32(((dim1 >> 16) & 0xFFFFu) | (tile0 << 16));    // dim1 hi | tile0
  g1[4] = (int)rfl_u32(tile1 & 0xFFFFu);                             // tile1 | tile2=0
  g1[5] = (int)rfl_u32(dim0);                                        // stride0 lo32
  g1[6] = 0;
  g1[7] = 0;
  v4i g2 = {0, 0, 0, 0};
  v4i g3 = {0, 0, 0, 0};
#if defined(__clang_major__) && (__clang_major__ >= 23)
  v8i g1x = {0, 0, 0, 0, 0, 0, 0, 0};
  __builtin_amdgcn_tensor_load_to_lds(g0, g1, g2, g3, g1x, 0);
#else
  __builtin_amdgcn_tensor_load_to_lds(g0, g1, g2, g3, 0);
#endif
  __builtin_amdgcn_s_wait_tensorcnt(0);
}
#endif

// ---------------------------------------------------------------------------
// Kernel A: peaks. One block per (b,k). LDS layout (dynamic):
//   float hm[25600]                        102400 B
//   u64   cand[CAND_CAP]                    65536 B
//   u64   rkey[256]                          2048 B
//   int   rslot[256]                         1024 B
//   int   counters[4]                          16 B
// ---------------------------------------------------------------------------
#define SMEM_A_BYTES ((size_t)HW * 4 + (size_t)CAND_CAP * 8 + 256 * 8 + 256 * 4 + 16)

__global__ void __launch_bounds__(256)
peaks_kernel(const float* __restrict__ heat, float* __restrict__ out_peaks) {
  extern __shared__ unsigned char smem[];
  float*              hm    = (float*)smem;
  unsigned long long* cand  = (unsigned long long*)(smem + (size_t)HW * 4);
  unsigned long long* rkey  = (unsigned long long*)(smem + (size_t)HW * 4 + (size_t)CAND_CAP * 8);
  int*                rslot = (int*)(smem + (size_t)HW * 4 + (size_t)CAND_CAP * 8 + 256 * 8);
  int*                ctr   = (int*)(smem + (size_t)HW * 4 + (size_t)CAND_CAP * 8 + 256 * 8 + 256 * 4);

  const int tid = threadIdx.x;
  const int bk  = blockIdx.x;                     // b*17 + k
  const float* __restrict__ src = heat + (size_t)bk * HW;

  if (tid == 0) ctr[0] = 0;

  // ---- Stage full 160x160 f32 heatmap into LDS --------------------------
#if HAVE_TDM
  if (tid < 32) {                                 // wave 0 issues one TDM op
    tdm_load_2d(src, (unsigned int)(uintptr_t)hm, 160u, 160u, 160u, 160u);
  }
#else
  {
    const float4* s4 = (const float4*)src;
    float4*       d4 = (float4*)hm;
    for (int p = tid; p < HW / 4; p += 256) d4[p] = s4[p];
  }
#endif
  __syncthreads();

  // ---- 3x3 NMS peak detection, collect candidates -----------------------
  for (int p = tid; p < HW; p += 256) {
    const int y = p / WW;
    const int x = p - y * WW;
    const float v = hm[p];
    bool peak = v > PEAK_TH;
    if (peak) {
      const int x0 = x > 0 ? x - 1 : 0, x1 = x < WW - 1 ? x + 1 : WW - 1;
      const int y0 = y > 0 ? y - 1 : 0, y1 = y < HH - 1 ? y + 1 : HH - 1;
      float m = v;
      for (int yy = y0; yy <= y1; ++yy)
        for (int xx = x0; xx <= x1; ++xx)
          m = fmaxf(m, hm[yy * WW + xx]);
      peak = (v == m);                             // v == pooled
    }
    if (peak) {
      int slot = atomicAdd(&ctr[0], 1);
      if (slot < CAND_CAP) {
        // key: descending value, then ascending pixel index (lax.top_k stable)
        cand[slot] = ((unsigned long long)__float_as_uint(v) << 32) |
                     (unsigned long long)(0xFFFFFFFFu - (unsigned int)p);
      }
    }
  }
  __syncthreads();

  const int ncand = ctr[0] < CAND_CAP ? ctr[0] : CAND_CAP;

  // ---- iterative argmax: select top-30 ----------------------------------
  for (int r = 0; r < PP; ++r) {
    unsigned long long bestK = 0ull;
    int bestS = -1;
    for (int c = tid; c < ncand; c += 256) {
      unsigned long long k = cand[c];
      if (k > bestK) { bestK = k; bestS = c; }
    }
    rkey[tid] = bestK;
    rslot[tid] = bestS;
    __syncthreads();
    for (int off = 128; off > 0; off >>= 1) {
      if (tid < off) {
        if (rkey[tid + off] > rkey[tid]) {
          rkey[tid] = rkey[tid + off];
          rslot[tid] = rslot[tid + off];
        }
      }
      __syncthreads();
    }
    if (tid == 0) {
      unsigned long long k = rkey[0];
      float px = 0.0f, py = 0.0f, sc = 0.0f;
      if (k != 0ull) {
        cand[rslot[0]] = 0ull;                     // remove winner
        const unsigned int pix = 0xFFFFFFFFu - (unsigned int)(k & 0xFFFFFFFFull);
        const float v = __uint_as_float((unsigned int)(k >> 32));
        const int yy = (int)pix / WW;
        const int xx = (int)pix - yy * WW;
        float dx = 0.0f, dy = 0.0f;
        if (xx > 0 && xx < WW - 1 && yy > 0 && yy < HH - 1) {
          const float rr = hm[pix + 1],  ll = hm[pix - 1];
          const float dd = hm[pix + WW], uu = hm[pix - WW];
          const float dxr = 0.5f * (rr - ll);
          const float dyr = 0.5f * (dd - uu);
          const float dxx = rr + ll - 2.0f * v;
          const float dyy = dd + uu - 2.0f * v;
          dx = (fabsf(dxx) > 1e-6f) ? dxr / (-dxx) : dxr;
          dy = (fabsf(dyy) > 1e-6f) ? dyr / (-dyy) : dyr;
        }
        px = (float)xx + dx;
        py = (float)yy + dy;
        sc = v;                                    // v > 0.1 => valid
      }
      float* o = out_peaks + ((size_t)bk * PP + r) * 3;
      o[0] = px; o[1] = py; o[2] = sc;
    }
    __syncthreads();
  }
}

// ---------------------------------------------------------------------------
// Kernel B: connections. One block per (b,l). 900 pairs x 10 samples.
// Both PAF channels (x then y, contiguous 51200 floats) live in LDS.
// 16 pairs per wave-group; per-pair sums via V_WMMA_F32_16X16X4_F32 with
// B = const (1/S for mean, 1 for count) so the B fragment layout is moot.
// LDS layout (dynamic): float pafl[51200] (204800 B) + 6*float[PP] (720 B).
// ---------------------------------------------------------------------------
#define SMEM_B_BYTES ((size_t)2 * HW * 4 + (size_t)6 * PP * 4)

// Branch-free LDS sampler: clamped sample index + arithmetic mask (keeps EXEC
// untouched around the WMMA region; no saveexec blocks).
__device__ __forceinline__ float paf_sample_lds(const float* __restrict__ pafl,
                                                float axi, float ayi,
                                                float dxl, float dyl,
                                                float vxn, float vyn,
                                                int s, float mask) {
  const int  se = s < SS ? s : SS - 1;
  const float t  = (float)se * (1.0f / 9.0f);     // linspace(0,1,10)
  const float xs = axi + t * dxl;
  const float ys = ayi + t * dyl;
  const float rx = fminf(fmaxf(rintf(xs), 0.0f), (float)(WW - 1));  // round-half-even
  const float ry = fminf(fmaxf(rintf(ys), 0.0f), (float)(HH - 1));
  const int lin = (int)ry * WW + (int)rx;
  const float v = pafl[lin] * vxn + pafl[HW + lin] * vyn;
  const float m = (s < SS) ? mask : 0.0f;
  return v * m;
}

__global__ void __launch_bounds__(256)
conn_kernel(const float* __restrict__ paf, const float* __restrict__ peaks,
            float* __restrict__ conn) {
  extern __shared__ unsigned char smemb[];
  float* pafl = (float*)smemb;                    // [2*HW]: x channel then y channel
  float* ax = (float*)(smemb + (size_t)2 * HW * 4);
  float* ay = ax + PP;
  float* sa = ay + PP;
  float* bx = sa + PP;
  float* by = bx + PP;
  float* sb = by + PP;

  const int tid = threadIdx.x;
  const int b   = blockIdx.x / LL;
  const int l   = blockIdx.x - b * LL;
  const int ka  = SK_A[l];
  const int kb  = SK_B[l];

  const float* __restrict__ pa = peaks + (size_t)(b * K_KP + ka) * PP * 3;
  const float* __restrict__ pb = peaks + (size_t)(b * K_KP + kb) * PP * 3;
  if (tid < PP) {
    ax[tid] = pa[tid * 3 + 0]; ay[tid] = pa[tid * 3 + 1]; sa[tid] = pa[tid * 3 + 2];
    bx[tid] = pb[tid * 3 + 0]; by[tid] = pb[tid * 3 + 1]; sb[tid] = pb[tid * 3 + 2];
  }

  // ---- Stage both PAF channels (contiguous 204800 B) into LDS -----------
  const float* __restrict__ pafg = paf + (size_t)(b * 38 + 2 * l) * HW;
#if HAVE_TDM
  if (tid < 32) {                                 // wave 0 issues one TDM op
    tdm_load_2d(pafg, (unsigned int)(uintptr_t)pafl, 2u * HW, 1u, 2u * HW, 1u);
  }
#else
  {
    const float4* s4 = (const float4*)pafg;
    float4*       d4 = (float4*)pafl;
    for (int p = tid; p < 2 * HW / 4; p += 256) d4[p] = s4[p];
  }
#endif
  __syncthreads();

  float* __restrict__ co = conn + (size_t)(b * LL + l) * (PP * PP);

  const int lane  = tid & 31;
  const int wave  = tid >> 5;
  const int nwave = 256 >> 5;
  const int row   = lane & 15;                    // A-fragment row M
  const int hi    = lane >> 4;                    // 0: K={0,1}, 1: K={2,3}

  const int nGroups = (PP * PP + 15) / 16;        // 57
  for (int g = wave; g < nGroups; g += nwave) {   // wave-uniform loop: EXEC full at WMMA
    const int pairIdx = g * 16 + row;
    const int cp   = pairIdx < PP * PP ? pairIdx : PP * PP - 1;
    const float live = pairIdx < PP * PP ? 1.0f : 0.0f;
    const int i = cp / PP;
    const int j = cp - i * PP;

    const float axi = ax[i], ayi = ay[i];
    const float dxl = bx[j] - axi;
    const float dyl = by[j] - ayi;
    const float nrm = sqrtf(dxl * dxl + dyl * dyl) + 1e-8f;
    const float vxn = dxl / nrm;
    const float vyn = dyl / nrm;

    v8f cm = {0, 0, 0, 0, 0, 0, 0, 0};            // mean accumulator (D/C 16x16 f32)
    v8f cc = {0, 0, 0, 0, 0, 0, 0, 0};            // count accumulator
    v2f bMean; bMean.x = 1.0f / (float)SS; bMean.y = 1.0f / (float)SS;
    v2f bOne;  bOne.x  = 1.0f;             bOne.y  = 1.0f;

#pragma unroll
    for (int c = 0; c < 3; ++c) {                 // K chunks: samples 4c..4c+3
      const int s0 = c * 4 + hi * 2;
      const float v0 = paf_sample_lds(pafl, axi, ayi, dxl, dyl, vxn, vyn, s0,     live);
      const float v1 = paf_sample_lds(pafl, axi, ayi, dxl, dyl, vxn, vyn, s0 + 1, live);
      v2f aM; aM.x = v0; aM.y = v1;
      v2f aC; aC.x = (v0 > PAF_TH) ? 1.0f : 0.0f;
              aC.y = (v1 > PAF_TH) ? 1.0f : 0.0f;
      cm = __builtin_amdgcn_wmma_f32_16x16x4_f32(false, aM, false, bMean, (short)0, cm, false, false);
      cc = __builtin_amdgcn_wmma_f32_16x16x4_f32(false, aC, false, bOne,  (short)0, cc, false, false);
    }

    // D layout: VGPR r holds M=r (lanes 0-15) / M=8+r (lanes 16-31), replicated over N.
    // Lane 0 emits rows 0-7, lane 16 emits rows 8-15 (column N=0).
    if ((lane & 15) == 0) {
#pragma unroll
      for (int r = 0; r < 8; ++r) {
        const int po = g * 16 + hi * 8 + r;
        if (po < PP * PP) {
          const int ii = po / PP;
          const int jj = po - ii * PP;
          const float mean  = cm[r];               // (1/S) * sum_s vecs
          const float count = cc[r];               // #(vecs > 0.05), exact small int
          const bool cond = (mean > 0.0f) && (count > 8.0f) &&     // ratio > 0.8
                            (sa[ii] > 0.0f) && (sb[jj] > 0.0f);    // pair_valid
          co[po] = cond ? (mean + 0.5f * (sa[ii] + sb[jj])) : 0.0f;
        }
      }
    }
  }
}

// ---------------------------------------------------------------------------
extern "C" void kernel_launch(void* const* d_in, const int* in_sizes, int n_in,
                              void* d_out, int out_size, void* d_ws, size_t ws_size,
                              hipStream_t stream) {
  (void)in_sizes; (void)n_in; (void)out_size; (void)d_ws; (void)ws_size;

  const float* heat = (const float*)d_in[0];      // (32,17,160,160) f32
  const float* paf  = (const float*)d_in[1];      // (32,38,160,160) f32
  float* out   = (float*)d_out;
  float* peaks = out;                              // 32*17*30*3 = 48960
  float* conn  = out + (size_t)B_SZ * K_KP * PP * 3;

  // 320KB-LDS WGP: opt in to large dynamic shared (no-op if already allowed).
  (void)hipFuncSetAttribute(reinterpret_cast<const void*>(&peaks_kernel),
                            hipFuncAttributeMaxDynamicSharedMemorySize,
                            (int)SMEM_A_BYTES);
  (void)hipFuncSetAttribute(reinterpret_cast<const void*>(&conn_kernel),
                            hipFuncAttributeMaxDynamicSharedMemorySize,
                            (int)SMEM_B_BYTES);

  peaks_kernel<<<B_SZ * K_KP, 256, SMEM_A_BYTES, stream>>>(heat, peaks);
  conn_kernel<<<B_SZ * LL, 256, SMEM_B_BYTES, stream>>>(paf, peaks, conn);
}